// ModelNew_3556232921949
// MI455X (gfx1250) — compile-verified
//
#include <hip/hip_runtime.h>

typedef __attribute__((ext_vector_type(2))) float v2f;
typedef __attribute__((ext_vector_type(8))) float v8f;

#define N_INPUT   4096   // columns of weight, inner dim
#define N_HIDDEN  4096   // rows of weight
#define BATCH     4096
#define ROW_SPLITS 32    // split-K for the column-sum stage
#define OUT_SCALE 1.0f   // SCALE_DIV * SCALING_FACTOR = 2.0 * 0.5

// ---------------------------------------------------------------------------
// Kernel 1: partial column sums of weight.
//   partial[s][col] = sum over rows [s*128, (s+1)*128) of weight[row][col]
// grid (N_INPUT/256, ROW_SPLITS), block 256. Fully coalesced: 256 consecutive
// threads read 1 KB contiguous per row step.
// ---------------------------------------------------------------------------
__global__ void colsum_partial_kernel(const float* __restrict__ w,
                                      float* __restrict__ partial) {
    const int col = blockIdx.x * 256 + threadIdx.x;
    const int rows_per_split = N_HIDDEN / ROW_SPLITS;
    const int r0 = blockIdx.y * rows_per_split;
    const float* p = w + (size_t)r0 * N_INPUT + col;
    float s = 0.0f;
#pragma unroll 8
    for (int r = 0; r < rows_per_split; ++r) {
        s += p[(size_t)r * N_INPUT];
    }
    partial[(size_t)blockIdx.y * N_INPUT + col] = s;
}

// ---------------------------------------------------------------------------
// Kernel 2: reduce the ROW_SPLITS partials into wsum[col].
// grid N_INPUT/256, block 256. Fixed summation order -> deterministic.
// ---------------------------------------------------------------------------
__global__ void colsum_final_kernel(const float* __restrict__ partial,
                                    float* __restrict__ wsum) {
    const int col = blockIdx.x * 256 + threadIdx.x;
    float s = 0.0f;
#pragma unroll
    for (int r = 0; r < ROW_SPLITS; ++r) {
        s += partial[(size_t)r * N_INPUT + col];
    }
    wsum[col] = s;
}

// ---------------------------------------------------------------------------
// Kernel 3: out[b] = OUT_SCALE * (x[b,:] . wsum) via V_WMMA_F32_16X16X4_F32.
// One wave (32 lanes) per 16-row tile of x; B-matrix columns are all wsum,
// synthesized in registers. Accumulate over K=4096 in steps of 4.
//
// A-fragment (16x4 f32, 2 VGPRs per ISA 7.12.2):
//   a.x : lanes 0-15 -> A[m, k+0], lanes 16-31 -> A[m, k+2]
//   a.y : lanes 0-15 -> A[m, k+1], lanes 16-31 -> A[m, k+3]
// B-fragment mirrored (rows striped the same way); since every column of B is
// identical, only the K-row mapping matters and it matches A's.
// C/D (16x16 f32, 8 VGPRs): VGPR r, lanes 0-15 -> M=r, N=lane;
//   lanes 16-31 -> M=8+r, N=lane-16.  All N columns hold the same dot product,
//   so lane 0 (M=0..7) and lane 16 (M=8..15) write the tile's outputs.
// ---------------------------------------------------------------------------
__global__ void gemv_wmma_kernel(const float* __restrict__ x,
                                 const float* __restrict__ wsum,
                                 float* __restrict__ out) {
    const int lane = threadIdx.x;        // blockDim.x == 32, full wave active
    const int m    = lane & 15;          // row within the 16-row tile
    const int h    = lane >> 4;          // K half-select (0 or 1)
    const int row0 = blockIdx.x * 16;

    const float* xrow = x + (size_t)(row0 + m) * N_INPUT;

    v8f acc = {};
#pragma unroll 8
    for (int k = 0; k < N_INPUT; k += 4) {
        const float2 xa = *(const float2*)(xrow + k + 2 * h);   // A[m, k+2h .. +1]
        const float2 wb = *(const float2*)(wsum + k + 2 * h);   // B rows k+2h .. +1
        v2f a; a.x = xa.x; a.y = xa.y;
        v2f b; b.x = wb.x; b.y = wb.y;
        // 8-arg form: (neg_a, A, neg_b, B, c_mod, C, reuse_a, reuse_b)
        acc = __builtin_amdgcn_wmma_f32_16x16x4_f32(
            false, a, false, b, (short)0, acc, false, false);
    }

    // Lanes 0 and 16 hold column N=0 of the result tile.
    if (m == 0) {
        const int mbase = row0 + (h ? 8 : 0);
#pragma unroll
        for (int r = 0; r < 8; ++r) {
            out[mbase + r] = acc[r] * OUT_SCALE;
        }
    }
}

// ---------------------------------------------------------------------------
// Launch: inputs are (x [BATCH*N_INPUT] f32, weight [N_HIDDEN*N_INPUT] f32),
// output is [BATCH] f32. Workspace: 32*4096 partials + 4096 wsum floats.
// ---------------------------------------------------------------------------
extern "C" void kernel_launch(void* const* d_in, const int* in_sizes, int n_in,
                              void* d_out, int out_size, void* d_ws, size_t ws_size,
                              hipStream_t stream) {
    const float* x = (const float*)d_in[0];
    const float* w = (const float*)d_in[1];
    float* out     = (float*)d_out;

    float* partial = (float*)d_ws;                         // ROW_SPLITS * N_INPUT floats
    float* wsum    = partial + (size_t)ROW_SPLITS * N_INPUT; // N_INPUT floats

    dim3 g1(N_INPUT / 256, ROW_SPLITS);
    colsum_partial_kernel<<<g1, 256, 0, stream>>>(w, partial);
    colsum_final_kernel<<<N_INPUT / 256, 256, 0, stream>>>(partial, wsum);
    gemv_wmma_kernel<<<BATCH / 16, 32, 0, stream>>>(x, wsum, out);
}